// GATConvolution_55662776156459
// MI455X (gfx1250) — compile-verified
//
#include <hip/hip_runtime.h>
#include <cfloat>
#include <cmath>

#define NNODES 50000
#define NEDGES 800000
#define ETOT   (NEDGES + NNODES)
#define FDIM   128
#define HEADS  4
#define CH     32
#define EDIM   16
#define NCLS   10
#define SLOPE  0.2f

typedef float v2f __attribute__((ext_vector_type(2)));
typedef float v8f __attribute__((ext_vector_type(8)));

static inline int cdiv(long long a, long long b) { return (int)((a + b - 1) / b); }

// ---------------- utility kernels ----------------
__global__ void fill_f32(float* p, float v, int n) {
    int i = blockIdx.x * blockDim.x + threadIdx.x;
    if (i < n) p[i] = v;
}

__device__ inline void atomicMaxF(float* addr, float v) {
    // works across mixed signs given init = -FLT_MAX
    if (v >= 0.f) atomicMax((int*)addr, __float_as_int(v));
    else          atomicMin((unsigned int*)addr, __float_as_uint(v));
}

// deg + edge_attr segment sums over dst (thread per (edge, dim))
__global__ void edge_deg_attr(const int* __restrict__ ei, const float* __restrict__ ea,
                              float* __restrict__ deg, float* __restrict__ asum) {
    int t = blockIdx.x * blockDim.x + threadIdx.x;
    if (t >= NEDGES * EDIM) return;
    int e = t >> 4, d = t & 15;
    int dst = ei[NEDGES + e];
    atomicAdd(&asum[dst * EDIM + d], ea[t]);
    if (d == 0) atomicAdd(&deg[dst], 1.0f);
}

// loop_attr = asum / max(deg,1)   (in place)
__global__ void loop_attr_fin(float* __restrict__ asum, const float* __restrict__ deg) {
    int t = blockIdx.x * blockDim.x + threadIdx.x;
    if (t >= NNODES * EDIM) return;
    asum[t] /= fmaxf(deg[t >> 4], 1.0f);
}

// M[d,h] = sum_c We[h*CH+c, d] * att_e[h,c]   (collapses ep @ att_e)
__global__ void build_M(const float* __restrict__ We, const float* __restrict__ att_e,
                        float* __restrict__ M) {
    int t = threadIdx.x;
    if (t >= EDIM * HEADS) return;
    int d = t / HEADS, h = t % HEADS;
    float s = 0.f;
    #pragma unroll
    for (int c = 0; c < CH; ++c) s += We[(h * CH + c) * EDIM + d] * att_e[h * CH + c];
    M[d * HEADS + h] = s;
}

// ---------------- WMMA GEMM: Y[N,128] = X[N,128] @ W[128,128]^T ----------------
// One block = 8 waves; block handles a 16-row stripe, wave w handles cols 16w..16w+15.
// W staged in LDS as [n][k] with stride 130 (bank-conflict pad).
__global__ __launch_bounds__(256) void gemm_xW_wmma(const float* __restrict__ X,
                                                    const float* __restrict__ W,
                                                    float* __restrict__ Y) {
    __shared__ float Wl[128 * 130];
    int tid = threadIdx.x;
    for (int i = tid; i < 128 * 128; i += 256)
        Wl[(i >> 7) * 130 + (i & 127)] = W[i];
    __syncthreads();

    int wave = tid >> 5, lane = tid & 31;
    int half = lane >> 4, lm = lane & 15;
    int n = wave * 16 + lm;                      // output column (B N-index)
    size_t row = (size_t)blockIdx.x * 16 + lm;   // A row (M-index)
    const float* xr = X + row * FDIM;
    const float* wr = Wl + n * 130;

    v8f acc = {};
    #pragma unroll
    for (int kb = 0; kb < FDIM; kb += 4) {
        int ka = kb + half * 2;                  // lanes 0-15: K=kb,kb+1; 16-31: kb+2,kb+3
        v2f a; a.x = xr[ka]; a.y = xr[ka + 1];
        v2f b; b.x = wr[ka]; b.y = wr[ka + 1];   // B[k,n] = W[n,k]
        acc = __builtin_amdgcn_wmma_f32_16x16x4_f32(
            false, a, false, b, (short)0, acc, false, false);
    }
    size_t r0 = (size_t)blockIdx.x * 16 + half * 8;
    #pragma unroll
    for (int r = 0; r < 8; ++r)
        Y[(r0 + r) * FDIM + n] = acc[r];         // C/D: vgpr r -> M = half*8 + r
}

// a_s / a_d per (node, head)
__global__ void attn_sd(const float* __restrict__ xp, const float* __restrict__ att_s,
                        const float* __restrict__ att_d, float* __restrict__ asb,
                        float* __restrict__ adb) {
    int t = blockIdx.x * blockDim.x + threadIdx.x;
    if (t >= NNODES * HEADS) return;
    int nidx = t >> 2, h = t & 3;
    const float* row = xp + (size_t)nidx * FDIM + h * CH;
    float ss = 0.f, sd = 0.f;
    #pragma unroll
    for (int c = 0; c < CH; ++c) {
        float v = row[c];
        ss += v * att_s[h * CH + c];
        sd += v * att_d[h * CH + c];
    }
    asb[t] = ss; adb[t] = sd;
}

// alpha = leaky_relu(a_s[src] + a_d[dst] + ea@M);  segment max into amax[dst]
__global__ void alpha_max(const int* __restrict__ ei, const float* __restrict__ ea,
                          const float* __restrict__ lattr, const float* __restrict__ M,
                          const float* __restrict__ asb, const float* __restrict__ adb,
                          float* __restrict__ alpha, float* __restrict__ amax) {
    int t = blockIdx.x * blockDim.x + threadIdx.x;
    if (t >= ETOT * HEADS) return;
    int e = t >> 2, h = t & 3;
    int src, dst; const float* arow;
    if (e < NEDGES) { src = ei[e]; dst = ei[NEDGES + e]; arow = ea + (size_t)e * EDIM; }
    else            { src = dst = e - NEDGES;            arow = lattr + (size_t)src * EDIM; }
    float ae = 0.f;
    #pragma unroll
    for (int d = 0; d < EDIM; ++d) ae += arow[d] * M[d * HEADS + h];
    float a = asb[src * HEADS + h] + adb[dst * HEADS + h] + ae;
    a = (a > 0.f) ? a : SLOPE * a;
    alpha[t] = a;
    atomicMaxF(&amax[dst * HEADS + h], a);
}

// ex = exp(alpha - amax[dst]); den[dst] += ex   (alpha buffer rewritten with ex)
__global__ void exp_den(const int* __restrict__ ei, float* __restrict__ alpha,
                        const float* __restrict__ amax, float* __restrict__ den) {
    int t = blockIdx.x * blockDim.x + threadIdx.x;
    if (t >= ETOT * HEADS) return;
    int e = t >> 2, h = t & 3;
    int dst = (e < NEDGES) ? ei[NEDGES + e] : (e - NEDGES);
    float ex = expf(alpha[t] - amax[dst * HEADS + h]);
    alpha[t] = ex;
    atomicAdd(&den[dst * HEADS + h], ex);
}

__global__ void init_bias(float* __restrict__ agg, const float* __restrict__ b) {
    int t = blockIdx.x * blockDim.x + threadIdx.x;
    if (t >= NNODES * FDIM) return;
    agg[t] = b[t & (FDIM - 1)];
}

// wave per edge: agg[dst] += xp[src] * (ex/den[dst]) over 128 channels (4/lane)
__global__ __launch_bounds__(256) void aggregate(const int* __restrict__ ei,
                                                 const float* __restrict__ xp,
                                                 const float* __restrict__ exw,
                                                 const float* __restrict__ den,
                                                 float* __restrict__ agg) {
    int g = blockIdx.x * blockDim.x + threadIdx.x;
    int e = g >> 5;
    if (e >= ETOT) return;
    int lane = g & 31;
    int src, dst;
    if (e < NEDGES) { src = ei[e]; dst = ei[NEDGES + e]; }
    else            { src = dst = e - NEDGES; }
    int c0 = lane << 2;             // 4 consecutive channels, all in same head
    int h = c0 >> 5;
    float w = exw[e * HEADS + h] / den[dst * HEADS + h];
    const float4 v = *reinterpret_cast<const float4*>(xp + (size_t)src * FDIM + c0);
    float* o = agg + (size_t)dst * FDIM + c0;
    atomicAdd(o + 0, v.x * w);
    atomicAdd(o + 1, v.y * w);
    atomicAdd(o + 2, v.z * w);
    atomicAdd(o + 3, v.w * w);
}

__global__ void elu_store(const float* __restrict__ agg, float* __restrict__ h1) {
    int t = blockIdx.x * blockDim.x + threadIdx.x;
    if (t >= NNODES * FDIM) return;
    float v = agg[t];
    h1[t] = (v > 0.f) ? v : (expf(v) - 1.f);
}

__global__ void elu_res(float* __restrict__ agg, const float* __restrict__ h1) {
    int t = blockIdx.x * blockDim.x + threadIdx.x;
    if (t >= NNODES * FDIM) return;
    float v = agg[t];
    agg[t] = ((v > 0.f) ? v : (expf(v) - 1.f)) + h1[t];
}

// logits = h2 @ Wl^T + bl, then log_softmax per row
__global__ void head_logsoftmax(const float* __restrict__ h2, const float* __restrict__ Wl,
                                const float* __restrict__ bl, float* __restrict__ out) {
    int n = blockIdx.x * blockDim.x + threadIdx.x;
    if (n >= NNODES) return;
    float acc[NCLS];
    #pragma unroll
    for (int c = 0; c < NCLS; ++c) acc[c] = bl[c];
    const float* hr = h2 + (size_t)n * FDIM;
    for (int k = 0; k < FDIM; ++k) {
        float hv = hr[k];
        #pragma unroll
        for (int c = 0; c < NCLS; ++c) acc[c] += hv * Wl[c * FDIM + k];
    }
    float m = acc[0];
    #pragma unroll
    for (int c = 1; c < NCLS; ++c) m = fmaxf(m, acc[c]);
    float s = 0.f;
    #pragma unroll
    for (int c = 0; c < NCLS; ++c) s += expf(acc[c] - m);
    float lse = m + logf(s);
    #pragma unroll
    for (int c = 0; c < NCLS; ++c) out[(size_t)n * NCLS + c] = acc[c] - lse;
}

// ---------------- driver ----------------
extern "C" void kernel_launch(void* const* d_in, const int* in_sizes, int n_in,
                              void* d_out, int out_size, void* d_ws, size_t ws_size,
                              hipStream_t stream) {
    const float* x   = (const float*)d_in[0];
    const int*   ei  = (const int*)  d_in[1];
    const float* ea  = (const float*)d_in[2];
    const float* W1  = (const float*)d_in[3];
    const float* We1 = (const float*)d_in[4];
    const float* as1 = (const float*)d_in[5];
    const float* ad1 = (const float*)d_in[6];
    const float* ae1 = (const float*)d_in[7];
    const float* b1  = (const float*)d_in[8];
    const float* W2  = (const float*)d_in[9];
    const float* We2 = (const float*)d_in[10];
    const float* as2 = (const float*)d_in[11];
    const float* ad2 = (const float*)d_in[12];
    const float* ae2 = (const float*)d_in[13];
    const float* b2  = (const float*)d_in[14];
    const float* Wlw = (const float*)d_in[15];
    const float* bl  = (const float*)d_in[16];
    float* out = (float*)d_out;

    // workspace layout (floats)
    float* ws = (float*)d_ws;
    size_t o = 0;
    float* xp    = ws + o; o += (size_t)NNODES * FDIM;   // projected features (reused L1/L2)
    float* h1    = ws + o; o += (size_t)NNODES * FDIM;   // elu(layer1)
    float* agg   = ws + o; o += (size_t)NNODES * FDIM;   // aggregation accumulator
    float* alpha = ws + o; o += (size_t)ETOT * HEADS;    // alpha / ex
    float* asb   = ws + o; o += (size_t)NNODES * HEADS;
    float* adb   = ws + o; o += (size_t)NNODES * HEADS;
    float* amax  = ws + o; o += (size_t)NNODES * HEADS;
    float* den   = ws + o; o += (size_t)NNODES * HEADS;
    float* deg   = ws + o; o += (size_t)NNODES;
    float* lattr = ws + o; o += (size_t)NNODES * EDIM;   // attr sums -> loop_attr
    float* M1    = ws + o; o += 64;
    float* M2    = ws + o; o += 64;
    (void)in_sizes; (void)n_in; (void)out_size; (void)ws_size;

    const int B = 256;

    // self-loop attr (shared by both layers)
    (void)hipMemsetAsync(deg, 0, (size_t)NNODES * sizeof(float), stream);
    (void)hipMemsetAsync(lattr, 0, (size_t)NNODES * EDIM * sizeof(float), stream);
    edge_deg_attr<<<cdiv((long long)NEDGES * EDIM, B), B, 0, stream>>>(ei, ea, deg, lattr);
    loop_attr_fin<<<cdiv((long long)NNODES * EDIM, B), B, 0, stream>>>(lattr, deg);
    build_M<<<1, 64, 0, stream>>>(We1, ae1, M1);
    build_M<<<1, 64, 0, stream>>>(We2, ae2, M2);

    // ---- layer 1 ----
    gemm_xW_wmma<<<NNODES / 16, B, 0, stream>>>(x, W1, xp);
    attn_sd<<<cdiv((long long)NNODES * HEADS, B), B, 0, stream>>>(xp, as1, ad1, asb, adb);
    fill_f32<<<cdiv((long long)NNODES * HEADS, B), B, 0, stream>>>(amax, -FLT_MAX, NNODES * HEADS);
    (void)hipMemsetAsync(den, 0, (size_t)NNODES * HEADS * sizeof(float), stream);
    alpha_max<<<cdiv((long long)ETOT * HEADS, B), B, 0, stream>>>(ei, ea, lattr, M1, asb, adb, alpha, amax);
    exp_den<<<cdiv((long long)ETOT * HEADS, B), B, 0, stream>>>(ei, alpha, amax, den);
    init_bias<<<cdiv((long long)NNODES * FDIM, B), B, 0, stream>>>(agg, b1);
    aggregate<<<cdiv((long long)ETOT * 32, B), B, 0, stream>>>(ei, xp, alpha, den, agg);
    elu_store<<<cdiv((long long)NNODES * FDIM, B), B, 0, stream>>>(agg, h1);

    // ---- layer 2 ----
    gemm_xW_wmma<<<NNODES / 16, B, 0, stream>>>(h1, W2, xp);
    attn_sd<<<cdiv((long long)NNODES * HEADS, B), B, 0, stream>>>(xp, as2, ad2, asb, adb);
    fill_f32<<<cdiv((long long)NNODES * HEADS, B), B, 0, stream>>>(amax, -FLT_MAX, NNODES * HEADS);
    (void)hipMemsetAsync(den, 0, (size_t)NNODES * HEADS * sizeof(float), stream);
    alpha_max<<<cdiv((long long)ETOT * HEADS, B), B, 0, stream>>>(ei, ea, lattr, M2, asb, adb, alpha, amax);
    exp_den<<<cdiv((long long)ETOT * HEADS, B), B, 0, stream>>>(ei, alpha, amax, den);
    init_bias<<<cdiv((long long)NNODES * FDIM, B), B, 0, stream>>>(agg, b2);
    aggregate<<<cdiv((long long)ETOT * 32, B), B, 0, stream>>>(ei, xp, alpha, den, agg);
    elu_res<<<cdiv((long long)NNODES * FDIM, B), B, 0, stream>>>(agg, h1);  // agg = elu(agg)+h1

    // ---- head ----
    head_logsoftmax<<<cdiv(NNODES, B), B, 0, stream>>>(agg, Wlw, bl, out);
}